// SigmoidTopK_81423989998118
// MI455X (gfx1250) — compile-verified
//
#include <hip/hip_runtime.h>

#define N       4096
#define TPB     256
#define EPT     16      // elements per thread (N / TPB)
#define NITERS  48

typedef __attribute__((address_space(3))) unsigned char lds_byte_t;

// ---------------- wave32 reductions ----------------
__device__ __forceinline__ float wave_sum_f(float v) {
    v += __shfl_xor(v, 16, 32);
    v += __shfl_xor(v, 8, 32);
    v += __shfl_xor(v, 4, 32);
    v += __shfl_xor(v, 2, 32);
    v += __shfl_xor(v, 1, 32);
    return v;
}
__device__ __forceinline__ int wave_sum_i(int v) {
    v += __shfl_xor(v, 16, 32);
    v += __shfl_xor(v, 8, 32);
    v += __shfl_xor(v, 4, 32);
    v += __shfl_xor(v, 2, 32);
    v += __shfl_xor(v, 1, 32);
    return v;
}
__device__ __forceinline__ float wave_max_f(float v) {
    v = fmaxf(v, __shfl_xor(v, 16, 32));
    v = fmaxf(v, __shfl_xor(v, 8, 32));
    v = fmaxf(v, __shfl_xor(v, 4, 32));
    v = fmaxf(v, __shfl_xor(v, 2, 32));
    v = fmaxf(v, __shfl_xor(v, 1, 32));
    return v;
}
__device__ __forceinline__ float wave_min_f(float v) {
    v = fminf(v, __shfl_xor(v, 16, 32));
    v = fminf(v, __shfl_xor(v, 8, 32));
    v = fminf(v, __shfl_xor(v, 4, 32));
    v = fminf(v, __shfl_xor(v, 2, 32));
    v = fminf(v, __shfl_xor(v, 1, 32));
    return v;
}

// ---------------- block (8-wave) reductions; result uniform across block ----
__device__ __forceinline__ float block_sum_f(float v, float* red, int tid) {
    v = wave_sum_f(v);
    if ((tid & 31) == 0) red[tid >> 5] = v;
    __syncthreads();
    float s = ((red[0] + red[1]) + (red[2] + red[3])) +
              ((red[4] + red[5]) + (red[6] + red[7]));
    __syncthreads();
    return s;
}
__device__ __forceinline__ int block_sum_i(int v, int* red, int tid) {
    v = wave_sum_i(v);
    if ((tid & 31) == 0) red[tid >> 5] = v;
    __syncthreads();
    int s = ((red[0] + red[1]) + (red[2] + red[3])) +
            ((red[4] + red[5]) + (red[6] + red[7]));
    __syncthreads();
    return s;
}
__device__ __forceinline__ float block_max_f(float v, float* red, int tid) {
    v = wave_max_f(v);
    if ((tid & 31) == 0) red[tid >> 5] = v;
    __syncthreads();
    float s = fmaxf(fmaxf(fmaxf(red[0], red[1]), fmaxf(red[2], red[3])),
                    fmaxf(fmaxf(red[4], red[5]), fmaxf(red[6], red[7])));
    __syncthreads();
    return s;
}
__device__ __forceinline__ float block_min_f(float v, float* red, int tid) {
    v = wave_min_f(v);
    if ((tid & 31) == 0) red[tid >> 5] = v;
    __syncthreads();
    float s = fminf(fminf(fminf(red[0], red[1]), fminf(red[2], red[3])),
                    fminf(fminf(red[4], red[5]), fminf(red[6], red[7])));
    __syncthreads();
    return s;
}

__global__ void __launch_bounds__(TPB)
sigmoid_topk_hard_kernel(const float* __restrict__ logits,
                         const int*   __restrict__ kptr,
                         float*       __restrict__ out) {
    __shared__ float rowbuf[N];      // 16 KB staging of one row
    __shared__ float redf[8];
    __shared__ int   redi[8];
    __shared__ int   eq[TPB];

    const int tid = threadIdx.x;
    const int row = blockIdx.x;
    const int k   = kptr[0];
    const float kf = (float)k;

    const float* rowp = logits + (size_t)row * N;

    // ---- Stage row into LDS via gfx1250 async global->LDS DMA path ----
    unsigned ldsbase = (unsigned)(size_t)(lds_byte_t*)&rowbuf[0];
    #pragma unroll
    for (int c = 0; c < 4; ++c) {
        unsigned off = (unsigned)(tid * 16 + c * 4096);   // bytes
        unsigned lds_addr = ldsbase + off;
        asm volatile("global_load_async_to_lds_b128 %0, %1, %2"
                     :
                     : "v"(lds_addr), "v"(off), "s"(rowp)
                     : "memory");
    }
    asm volatile("s_wait_asynccnt 0" ::: "memory");
    __syncthreads();

    // ---- Pull this thread's contiguous 16-element slice into VGPRs ----
    float x[EPT];
    #pragma unroll
    for (int c = 0; c < 4; ++c) {
        float4 v = *reinterpret_cast<const float4*>(&rowbuf[tid * EPT + c * 4]);
        x[c * 4 + 0] = v.x; x[c * 4 + 1] = v.y;
        x[c * 4 + 2] = v.z; x[c * 4 + 3] = v.w;
    }

    // ---- Row max/min for bisection bounds ----
    float mx = -__builtin_inff(), mn = __builtin_inff();
    #pragma unroll
    for (int j = 0; j < EPT; ++j) { mx = fmaxf(mx, x[j]); mn = fminf(mn, x[j]); }
    mx = block_max_f(mx, redf, tid);
    mn = block_min_f(mn, redf, tid);

    // ---- Precompute e_j = exp(-x_j) once, and order-preserving uint keys ----
    float    e[EPT];
    unsigned key[EPT];
    #pragma unroll
    for (int j = 0; j < EPT; ++j) {
        e[j] = __expf(-x[j]);
        unsigned fu = __float_as_uint(x[j]);
        key[j] = (fu & 0x80000000u) ? ~fu : (fu | 0x80000000u);
    }

    // ---- 48-step bisection: sigmoid(x+mid) = 1 / (1 + e_j * exp(-mid)) ----
    float lo = -mx - 10.0f;
    float hi = -mn + 10.0f;
    for (int it = 0; it < NITERS; ++it) {
        float mid = 0.5f * (lo + hi);
        float w = __expf(-mid);                       // one transcendental/row/iter
        float acc = 0.0f;
        #pragma unroll
        for (int j = 0; j < EPT; ++j)
            acc += __builtin_amdgcn_rcpf(fmaf(e[j], w, 1.0f));
        float count = block_sum_f(acc, redf, tid);
        if (count < kf) lo = mid; else hi = mid;      // uniform across block
    }
    const float t = 0.5f * (lo + hi);

    // ---- Exact k-th-largest selection: 32-step radix bisection on keys ----
    unsigned klo = 0u, khi = 0xFFFFFFFFu;
    while (klo < khi) {                               // uniform condition
        unsigned mid = klo + ((khi - klo) >> 1) + 1u; // mid in (klo, khi]
        int c = 0;
        #pragma unroll
        for (int j = 0; j < EPT; ++j) c += (key[j] >= mid) ? 1 : 0;
        int cnt = block_sum_i(c, redi, tid);
        if (cnt >= k) klo = mid; else khi = mid - 1u;
    }
    const unsigned kth = klo;                          // k-th largest key value

    // ---- Tie bookkeeping: how many == kth must still be admitted ----
    int cg = 0, myeq = 0;
    #pragma unroll
    for (int j = 0; j < EPT; ++j) {
        cg   += (key[j] > kth)  ? 1 : 0;
        myeq += (key[j] == kth) ? 1 : 0;
    }
    const int count_gt  = block_sum_i(cg, redi, tid);
    const int remaining = k - count_gt;

    eq[tid] = myeq;
    __syncthreads();
    int eqrank = 0;                                    // exclusive prefix over threads
    for (int i = 0; i < tid; ++i) eqrank += eq[i];

    // ---- Emit: selected -> (1 - soft) + soft ; else -> (0 - soft) + soft == 0
    const float wf = __expf(-t);
    float* orow = out + (size_t)row * N;
    #pragma unroll
    for (int c = 0; c < 4; ++c) {
        float o[4];
        #pragma unroll
        for (int q = 0; q < 4; ++q) {
            int j = c * 4 + q;
            float soft = __builtin_amdgcn_rcpf(fmaf(e[j], wf, 1.0f));
            bool sel;
            if (key[j] > kth)       sel = true;
            else if (key[j] == kth) { sel = (eqrank < remaining); ++eqrank; }
            else                    sel = false;
            float oneh = sel ? 1.0f : 0.0f;
            float d = oneh - soft;                     // keep FP order: no reassoc
            o[q] = d + soft;                           // == 0 exactly when !sel
        }
        float4 v; v.x = o[0]; v.y = o[1]; v.z = o[2]; v.w = o[3];
        *reinterpret_cast<float4*>(&orow[tid * EPT + c * 4]) = v;
    }
}

extern "C" void kernel_launch(void* const* d_in, const int* in_sizes, int n_in,
                              void* d_out, int out_size, void* d_ws, size_t ws_size,
                              hipStream_t stream) {
    const float* logits = (const float*)d_in[0];
    const int*   kptr   = (const int*)d_in[1];
    float*       out    = (float*)d_out;
    const int rows = in_sizes[0] / N;                  // 8192 for the reference shapes
    sigmoid_topk_hard_kernel<<<dim3(rows), dim3(TPB), 0, stream>>>(logits, kptr, out);
}